// GraphConvolutionalNetwork_51797305590026
// MI455X (gfx1250) — compile-verified
//
#include <hip/hip_runtime.h>

#define NN 100000      // N_NODES
#define NE 1200000     // N_EDGES
#define NG 4096        // N_GRAPHS
#define INC 9
#define HID 64
#define NL 3
#define EPSV 1e-5f

typedef __attribute__((ext_vector_type(2))) float v2f;
typedef __attribute__((ext_vector_type(8))) float v8f;

// ---------------- utility ----------------
__global__ void zero_f(float* __restrict__ p, int n) {
  int i = blockIdx.x * blockDim.x + threadIdx.x;
  if (i < n) p[i] = 0.0f;
}

// deg[dst] += 1
__global__ void deg_count(const int* __restrict__ dst, float* __restrict__ deg) {
  int e = blockIdx.x * blockDim.x + threadIdx.x;
  if (e < NE) unsafeAtomicAdd(&deg[dst[e]], 1.0f);
}

// dis = rsqrt(deg + 1)   (self-loop included)
__global__ void deg_to_dis(float* __restrict__ deg) {
  int i = blockIdx.x * blockDim.x + threadIdx.x;
  if (i < NN) deg[i] = rsqrtf(deg[i] + 1.0f);
}

// ---------------- layer 0 input GEMM (K=9, scalar; negligible work) ----------------
__global__ void gemm_in(const float* __restrict__ x, const float* __restrict__ W0,
                        float* __restrict__ out) {
  int t = blockIdx.x * blockDim.x + threadIdx.x;
  if (t >= NN * HID) return;
  int n = t >> 6, c = t & 63;
  float s = 0.0f;
#pragma unroll
  for (int k = 0; k < INC; ++k) s = fmaf(x[n * INC + k], W0[k * HID + c], s);
  out[t] = s;
}

// ---------------- 64x64 GEMM via V_WMMA_F32_16X16X4_F32 ----------------
// One wave computes a 16x64 output tile; W (64x64 fp32) staged in LDS.
__global__ void gemm64_wmma(const float* __restrict__ A, const float* __restrict__ Wg,
                            float* __restrict__ out) {
  __shared__ float sW[HID * HID];
  for (int i = threadIdx.x; i < HID * HID; i += blockDim.x) sW[i] = Wg[i];
  __syncthreads();

  int wave = threadIdx.x >> 5;
  int tile = blockIdx.x * 8 + wave;           // 16 rows per tile; 6250 tiles total
  if (tile * 16 >= NN) return;                // wave-uniform exit -> EXEC all-ones for WMMA

  int lane = threadIdx.x & 31;
  int m    = lane & 15;                       // A: row M / B,C,D: col N
  int hi   = lane >> 4;                       // half-wave select
  int kl   = hi * 2;                          // K sub-offset for A/B fragments

  const float* arow = A + (size_t)(tile * 16 + m) * HID;

  v8f acc[4] = {};                            // 4 N-tiles of 16 cols -> 64 cols
  for (int k0 = 0; k0 < HID; k0 += 4) {
    v2f a;
    a.x = arow[k0 + kl];
    a.y = arow[k0 + kl + 1];
#pragma unroll
    for (int nt = 0; nt < 4; ++nt) {
      v2f b;
      b.x = sW[(k0 + kl)     * HID + nt * 16 + m];
      b.y = sW[(k0 + kl + 1) * HID + nt * 16 + m];
      acc[nt] = __builtin_amdgcn_wmma_f32_16x16x4_f32(
          /*neg_a=*/false, a, /*neg_b=*/false, b,
          /*c_mod=*/(short)0, acc[nt], /*reuse_a=*/false, /*reuse_b=*/false);
    }
  }
#pragma unroll
  for (int nt = 0; nt < 4; ++nt)
#pragma unroll
    for (int v = 0; v < 8; ++v)
      out[(size_t)(tile * 16 + v + 8 * hi) * HID + nt * 16 + m] = acc[nt][v];
}

// agg = hlin * dis^2 + b   (self-loop term; also initializes agg)
__global__ void self_bias(const float* __restrict__ hlin, const float* __restrict__ dis,
                          const float* __restrict__ b, float* __restrict__ agg) {
  int t = blockIdx.x * blockDim.x + threadIdx.x;
  if (t >= NN * HID) return;
  int n = t >> 6, c = t & 63;
  float d = dis[n];
  agg[t] = fmaf(hlin[t], d * d, b[c]);
}

// agg[dst] += hlin[src] * dis[src]*dis[dst]   (16 lanes/edge, float4 gather, fp32 atomics)
__global__ void edge_scatter(const int* __restrict__ src, const int* __restrict__ dstv,
                             const float* __restrict__ dis, const float* __restrict__ hlin,
                             float* __restrict__ agg) {
  unsigned t = blockIdx.x * blockDim.x + threadIdx.x;
  if (t >= (unsigned)NE * 16u) return;
  int e  = (int)(t >> 4);
  int c4 = (int)(t & 15u) * 4;
  int s = src[e], d = dstv[e];
  float nrm = dis[s] * dis[d];
  const float4 v = *(const float4*)(hlin + (size_t)s * HID + c4);
  float* base = agg + (size_t)d * HID + c4;
  unsafeAtomicAdd(base + 0, v.x * nrm);
  unsafeAtomicAdd(base + 1, v.y * nrm);
  unsafeAtomicAdd(base + 2, v.z * nrm);
  unsafeAtomicAdd(base + 3, v.w * nrm);
}

// per-column mean/biased-variance; one block (256 thr) per column
__global__ void bn_stats(const float* __restrict__ a, float* __restrict__ mu,
                         float* __restrict__ var) {
  __shared__ float ssum[256], ssq[256];
  int c = blockIdx.x;
  float s = 0.0f, q = 0.0f;
  for (int r = threadIdx.x; r < NN; r += 256) {
    float v = a[(size_t)r * HID + c];
    s += v; q = fmaf(v, v, q);
  }
  ssum[threadIdx.x] = s; ssq[threadIdx.x] = q;
  __syncthreads();
  for (int o = 128; o > 0; o >>= 1) {
    if (threadIdx.x < o) { ssum[threadIdx.x] += ssum[threadIdx.x + o];
                           ssq[threadIdx.x]  += ssq[threadIdx.x + o]; }
    __syncthreads();
  }
  if (threadIdx.x == 0) {
    float m = ssum[0] * (1.0f / NN);
    mu[c]  = m;
    var[c] = ssq[0] * (1.0f / NN) - m * m;
  }
}

// h = [h +] relu(g*(agg-mu)*rsqrt(var+eps)+be)
__global__ void bn_apply(const float* __restrict__ agg, const float* __restrict__ mu,
                         const float* __restrict__ var, const float* __restrict__ g,
                         const float* __restrict__ be, float* __restrict__ h, int residual) {
  int t = blockIdx.x * blockDim.x + threadIdx.x;
  if (t >= NN * HID) return;
  int c = t & 63;
  float y = fmaf(g[c] * (agg[t] - mu[c]), rsqrtf(var[c] + EPSV), be[c]);
  y = fmaxf(y, 0.0f);
  h[t] = residual ? (h[t] + y) : y;
}

// pooled[batch[n]] += h[n]; cnt[batch[n]] += 1
__global__ void pool_scatter(const int* __restrict__ batch, const float* __restrict__ h,
                             float* __restrict__ pooled, float* __restrict__ cnt) {
  unsigned t = blockIdx.x * blockDim.x + threadIdx.x;
  if (t >= (unsigned)NN * 16u) return;
  int n  = (int)(t >> 4);
  int c4 = (int)(t & 15u) * 4;
  int g = batch[n];
  const float4 v = *(const float4*)(h + (size_t)n * HID + c4);
  float* base = pooled + (size_t)g * HID + c4;
  unsafeAtomicAdd(base + 0, v.x);
  unsafeAtomicAdd(base + 1, v.y);
  unsafeAtomicAdd(base + 2, v.z);
  unsafeAtomicAdd(base + 3, v.w);
  if (c4 == 0) unsafeAtomicAdd(&cnt[g], 1.0f);
}

// per-graph MLP head: relu(pooled/cnt @ Wo1 + bo1) @ Wo2 + bo2
__global__ void head(const float* __restrict__ pooled, const float* __restrict__ cnt,
                     const float* __restrict__ Wo1, const float* __restrict__ bo1,
                     const float* __restrict__ Wo2, const float* __restrict__ bo2,
                     float* __restrict__ out) {
  int g = blockIdx.x * blockDim.x + threadIdx.x;
  if (g >= NG) return;
  float inv = 1.0f / fmaxf(cnt[g], 1.0f);
  float p[HID];
#pragma unroll
  for (int k = 0; k < HID; ++k) p[k] = pooled[(size_t)g * HID + k] * inv;
  float o = bo2[0];
  for (int j = 0; j < 32; ++j) {
    float z = bo1[j];
#pragma unroll
    for (int k = 0; k < HID; ++k) z = fmaf(p[k], Wo1[k * 32 + j], z);
    o = fmaf(fmaxf(z, 0.0f), Wo2[j], o);
  }
  out[g] = o;
}

extern "C" void kernel_launch(void* const* d_in, const int* in_sizes, int n_in,
                              void* d_out, int out_size, void* d_ws, size_t ws_size,
                              hipStream_t stream) {
  (void)in_sizes; (void)n_in; (void)out_size; (void)ws_size;
  const float* x    = (const float*)d_in[0];
  const int*   ei   = (const int*)d_in[1];          // [2, NE] (int32 under default JAX x64-off)
  const int*   bat  = (const int*)d_in[2];
  const float* W0   = (const float*)d_in[3];
  const float* b0   = (const float*)d_in[4];
  const float* g0   = (const float*)d_in[5];
  const float* be0  = (const float*)d_in[6];
  const float* Ws   = (const float*)d_in[7];        // [3,64,64]
  const float* bs   = (const float*)d_in[8];
  const float* gs   = (const float*)d_in[9];
  const float* bes  = (const float*)d_in[10];
  const float* Wo1  = (const float*)d_in[11];
  const float* bo1  = (const float*)d_in[12];
  const float* Wo2  = (const float*)d_in[13];
  const float* bo2  = (const float*)d_in[14];
  float* out = (float*)d_out;

  const int* src = ei;
  const int* dst = ei + NE;

  // workspace layout (floats)
  float* w = (float*)d_ws;
  float* dis    = w;                  w += NN;
  float* h      = w;                  w += (size_t)NN * HID;
  float* hlin   = w;                  w += (size_t)NN * HID;
  float* agg    = w;                  w += (size_t)NN * HID;
  float* mu     = w;                  w += HID;
  float* var    = w;                  w += HID;
  float* pooled = w;                  w += (size_t)NG * HID;   // cnt directly after
  float* cnt    = w;                  w += NG;

  const int TPB = 256;
  const int NH  = NN * HID;                 // 6,400,000
  const int gNH = (NH + TPB - 1) / TPB;
  const int gE16 = (int)(((size_t)NE * 16 + TPB - 1) / TPB);
  const int gN16 = (int)(((size_t)NN * 16 + TPB - 1) / TPB);
  const int gTiles = (NN / 16 + 7) / 8;     // 8 waves/block, 16 rows/wave

  // degree -> dis
  zero_f<<<(NN + TPB - 1) / TPB, TPB, 0, stream>>>(dis, NN);
  deg_count<<<(NE + TPB - 1) / TPB, TPB, 0, stream>>>(dst, dis);
  deg_to_dis<<<(NN + TPB - 1) / TPB, TPB, 0, stream>>>(dis);

  // layer 0
  gemm_in<<<gNH, TPB, 0, stream>>>(x, W0, hlin);
  self_bias<<<gNH, TPB, 0, stream>>>(hlin, dis, b0, agg);
  edge_scatter<<<gE16, TPB, 0, stream>>>(src, dst, dis, hlin, agg);
  bn_stats<<<HID, 256, 0, stream>>>(agg, mu, var);
  bn_apply<<<gNH, TPB, 0, stream>>>(agg, mu, var, g0, be0, h, 0);

  // hidden layers with residual
  for (int i = 0; i < NL; ++i) {
    gemm64_wmma<<<gTiles, TPB, 0, stream>>>(h, Ws + (size_t)i * HID * HID, hlin);
    self_bias<<<gNH, TPB, 0, stream>>>(hlin, dis, bs + i * HID, agg);
    edge_scatter<<<gE16, TPB, 0, stream>>>(src, dst, dis, hlin, agg);
    bn_stats<<<HID, 256, 0, stream>>>(agg, mu, var);
    bn_apply<<<gNH, TPB, 0, stream>>>(agg, mu, var, gs + i * HID, bes + i * HID, h, 1);
  }

  // global mean pool + head
  zero_f<<<(NG * HID + NG + TPB - 1) / TPB, TPB, 0, stream>>>(pooled, NG * HID + NG);
  pool_scatter<<<gN16, TPB, 0, stream>>>(bat, h, pooled, cnt);
  head<<<(NG + TPB - 1) / TPB, TPB, 0, stream>>>(pooled, cnt, Wo1, bo1, Wo2, bo2, out);
}